// IncrementalGraphConstructor_33990371180762
// MI455X (gfx1250) — compile-verified
//
#include <hip/hip_runtime.h>

#define BB    8
#define NN    2048
#define DIMM  25
#define NH    4
#define ODIM  32
#define DD    128      // NH*ODIM
#define UCOLS 1024     // only first k-chunk is used by the reference (zip truncation)
#define NEIGH 32

typedef float v2f __attribute__((ext_vector_type(2)));
typedef float v8f __attribute__((ext_vector_type(8)));

// ---------------------------------------------------------------------------
// Kernel 1: full per-node pipeline (fusion LN, LN, gate, nv) + Q/K projections.
// One thread per (b, n). Writes Pq/Pk as [b][n][128] (rows are WMMA A/B rows).
// ---------------------------------------------------------------------------
__global__ __launch_bounds__(256) void proj_kernel(
    const float* __restrict__ x,   const float* __restrict__ emb,
    const float* __restrict__ g,   const float* __restrict__ be,
    const float* __restrict__ w1,  const float* __restrict__ w2,
    const float* __restrict__ gb,  const float* __restrict__ wq,
    const float* __restrict__ bqv, const float* __restrict__ wk,
    const float* __restrict__ bkv, const int* __restrict__ increp,
    float* __restrict__ Pq, float* __restrict__ Pk)
{
  const int gid = blockIdx.x * blockDim.x + threadIdx.x;   // 0 .. B*N-1
  const int n   = gid & (NN - 1);
  const int inc = increp[0];

  const float* xr = x + (size_t)gid * DIMM;
  float xv[DIMM], gg[DIMM], bb[DIMM], t[DIMM];
#pragma unroll
  for (int c = 0; c < DIMM; ++c) { xv[c] = xr[c]; gg[c] = g[c]; bb[c] = be[c]; }

  if (inc) {
    // p2i = [x24, 0...0]; fusion = LN(p2i); inp = x with ch24 zeroed
    const float p2  = xv[24];
    const float m   = p2 * (1.0f / DIMM);
    const float var = ((p2 - m) * (p2 - m) + (DIMM - 1) * m * m) * (1.0f / DIMM);
    const float inv = rsqrtf(var + 1e-5f);
#pragma unroll
    for (int c = 0; c < DIMM; ++c) {
      const float p2i = (c == 0) ? p2 : 0.0f;
      const float fus = (p2i - m) * inv * gg[c] + bb[c];
      const float inp = (c < 24) ? xv[c] : 0.0f;
      t[c] = inp + fus;
    }
  } else {
#pragma unroll
    for (int c = 0; c < DIMM; ++c) t[c] = xv[c];
  }

  // node_input = LN(t)
  float mean = 0.0f;
#pragma unroll
  for (int c = 0; c < DIMM; ++c) mean += t[c];
  mean *= (1.0f / DIMM);
  float var = 0.0f;
#pragma unroll
  for (int c = 0; c < DIMM; ++c) { const float d = t[c] - mean; var += d * d; }
  var *= (1.0f / DIMM);
  const float inv = rsqrtf(var + 1e-5f);
  float ni[DIMM];
#pragma unroll
  for (int c = 0; c < DIMM; ++c) ni[c] = (t[c] - mean) * inv * gg[c] + bb[c];

  // gate / nv
  const float* st = emb + (size_t)n * DIMM;
  float stv[DIMM];
#pragma unroll
  for (int c = 0; c < DIMM; ++c) stv[c] = st[c];
  float nv[DIMM];
#pragma unroll 1
  for (int c = 0; c < DIMM; ++c) {
    float s = gb[c];
#pragma unroll
    for (int i = 0; i < DIMM; ++i)
      s += stv[i] * w1[i * DIMM + c] + ni[i] * w2[i * DIMM + c];
    const float gt = 1.0f / (1.0f + expf(-s));
    nv[c] = gt * stv[c] + (1.0f - gt) * ni[c] + stv[c];
  }

  // projections (25 x 128), bias included
  float* pq = Pq + (size_t)gid * DD;
  float* pk = Pk + (size_t)gid * DD;
#pragma unroll 1
  for (int j = 0; j < DD; ++j) {
    float aq = bqv[j], ak = bkv[j];
#pragma unroll
    for (int c = 0; c < DIMM; ++c) {
      aq += nv[c] * wq[c * DD + j];
      ak += nv[c] * wk[c * DD + j];
    }
    pq[j] = aq;
    pk[j] = ak;
  }
}

// ---------------------------------------------------------------------------
// Kernel 2: per (b, 16-row tile): scores[16][1024] in LDS via fp32 WMMA,
// then per-row top-32 + zero-fill + scatter into the 128 MB output.
// Block = 256 threads (8 waves); wave w owns u-tiles {w, w+8, ..., w+56}.
// B operands are batch-loaded into a register buffer per tile (one load
// clause + one wait), then 4 independent WMMA accumulation chains run
// back-to-back; the next tile's B row is prefetched (global_prefetch_b8).
// ---------------------------------------------------------------------------
__global__ __launch_bounds__(256) void attn_topk_kernel(
    const float* __restrict__ Pq, const float* __restrict__ Pk,
    const float* __restrict__ mlpw, const float* __restrict__ mlpb,
    float* __restrict__ out)
{
  __shared__ float sc[16 * UCOLS];   // 64 KB of the 320 KB WGP LDS

  const int b    = blockIdx.x >> 7;        // B * (N/16) blocks
  const int n0   = (blockIdx.x & 127) * 16;
  const int wave = threadIdx.x >> 5;
  const int lane = threadIdx.x & 31;
  const int l16  = lane & 15;
  const int half = lane >> 4;

  // head-mix weights (uniform; L0-cached broadcast loads)
  float w[16], mb[4];
#pragma unroll
  for (int i = 0; i < 16; ++i) w[i] = mlpw[i];
#pragma unroll
  for (int i = 0; i < 4; ++i)  mb[i] = mlpb[i];

  const float scale = 0.17677669529663688f;  // 1/sqrt(32)

  // A-matrix 16x4 f32 layout: M = lane%16 ; VGPR pair holds K = kb + 2*(lane/16) + {0,1}
  // Preload all 32 A float2s once (tile-invariant).
  const float* aRow = Pq + ((size_t)b * NN + n0 + l16) * DD + 2 * half;
  float2 afs[32];
#pragma unroll
  for (int i = 0; i < 32; ++i) afs[i] = *(const float2*)(aRow + i * 4);

  const float* bBase = Pk + ((size_t)b * NN + l16) * DD + 2 * half;

#pragma unroll 1
  for (int tile = 0; tile < 8; ++tile) {
    const int u0 = (wave + tile * 8) * 16;
    const float* bRow = bBase + (size_t)u0 * DD;

    // Batch-load all B operands for this tile: one clause of 32 b64 loads.
    float2 bfs[32];
#pragma unroll
    for (int i = 0; i < 32; ++i) bfs[i] = *(const float2*)(bRow + i * 4);

    // Prefetch next tile's B row (covers the 512B row with 4 lines).
    if (tile < 7) {
      const float* nb = bRow + (size_t)128 * DD;   // +8 u-tiles * 16 rows
      __builtin_prefetch(nb + 0,  0, 0);
      __builtin_prefetch(nb + 32, 0, 0);
      __builtin_prefetch(nb + 64, 0, 0);
      __builtin_prefetch(nb + 96, 0, 0);
    }

    v8f acc[NH];
#pragma unroll
    for (int h = 0; h < NH; ++h) acc[h] = (v8f){0.f,0.f,0.f,0.f,0.f,0.f,0.f,0.f};

    // 4 independent accumulation chains, 8 chained K-steps each.
#pragma unroll
    for (int j = 0; j < 8; ++j) {
#pragma unroll
      for (int h = 0; h < NH; ++h) {
        const float2 af = afs[h * 8 + j];
        const float2 bf = bfs[h * 8 + j];
        const v2f av = {af.x, af.y};
        const v2f bv = {bf.x, bf.y};
        acc[h] = __builtin_amdgcn_wmma_f32_16x16x4_f32(
            false, av, false, bv, (short)0, acc[h], false, false);
      }
    }

    // combine heads: score = sum_h [ relu(sum_i W[h][i]*a_i + b[h]) + a_h ]
#pragma unroll
    for (int r = 0; r < 8; ++r) {
      const float a0 = acc[0][r] * scale;
      const float a1 = acc[1][r] * scale;
      const float a2 = acc[2][r] * scale;
      const float a3 = acc[3][r] * scale;
      float s = a0 + a1 + a2 + a3;
      s += fmaxf(w[ 0]*a0 + w[ 1]*a1 + w[ 2]*a2 + w[ 3]*a3 + mb[0], 0.0f);
      s += fmaxf(w[ 4]*a0 + w[ 5]*a1 + w[ 6]*a2 + w[ 7]*a3 + mb[1], 0.0f);
      s += fmaxf(w[ 8]*a0 + w[ 9]*a1 + w[10]*a2 + w[11]*a3 + mb[2], 0.0f);
      s += fmaxf(w[12]*a0 + w[13]*a1 + w[14]*a2 + w[15]*a3 + mb[3], 0.0f);
      sc[(r + 8 * half) * UCOLS + u0 + l16] = s;   // C/D layout: M = r + 8*(lane/16)
    }
  }

  __syncthreads();

  // ---- top-32 + scatter: wave handles rows 2*wave and 2*wave+1 ----
#pragma unroll 1
  for (int rr = 0; rr < 2; ++rr) {
    const int row = wave * 2 + rr;
    float* outRow = out + ((size_t)b * NN + (n0 + row)) * NN;

    // zero-fill the full 2048-float output row (float4 stores)
    const float4 z4 = {0.f, 0.f, 0.f, 0.f};
    float4* p4 = (float4*)outRow;
#pragma unroll 1
    for (int i = lane; i < NN / 4; i += 32) p4[i] = z4;
    // commit zeros before the scatter stores below (same-wave ordering)
    asm volatile("s_wait_storecnt 0x0" ::: "memory");

    float v[32];
#pragma unroll
    for (int j = 0; j < 32; ++j) v[j] = sc[row * UCOLS + lane + 32 * j];

#pragma unroll 1
    for (int sel = 0; sel < NEIGH; ++sel) {
      float bv = v[0];
      int   bj = 0;
#pragma unroll
      for (int j = 1; j < 32; ++j)
        if (v[j] > bv) { bv = v[j]; bj = j; }          // strict > : smallest u per lane
      int bu = lane + (bj << 5);
#pragma unroll
      for (int off = 16; off > 0; off >>= 1) {          // wave32 argmax reduce
        const float ov = __shfl_xor(bv, off, 32);
        const int   ou = __shfl_xor(bu, off, 32);
        if (ov > bv || (ov == bv && ou < bu)) { bv = ov; bu = ou; }
      }
      if ((bu & 31) == lane) {                          // owner retires its value
        const int jw = bu >> 5;
#pragma unroll
        for (int j = 0; j < 32; ++j)
          if (j == jw) v[j] = -__builtin_inff();
      }
      if (lane == 0) outRow[bu] = bv;                   // scatter top value
    }
  }
}

// ---------------------------------------------------------------------------
extern "C" void kernel_launch(void* const* d_in, const int* in_sizes, int n_in,
                              void* d_out, int out_size, void* d_ws, size_t ws_size,
                              hipStream_t stream) {
  const float* x    = (const float*)d_in[0];
  const float* emb  = (const float*)d_in[1];
  const float* gam  = (const float*)d_in[2];
  const float* bet  = (const float*)d_in[3];
  const float* w1   = (const float*)d_in[4];
  const float* w2   = (const float*)d_in[5];
  const float* gb   = (const float*)d_in[6];
  const float* wq   = (const float*)d_in[7];
  const float* bq   = (const float*)d_in[8];
  const float* wk   = (const float*)d_in[9];
  const float* bk   = (const float*)d_in[10];
  const float* mlpw = (const float*)d_in[11];
  const float* mlpb = (const float*)d_in[12];
  const int*   inc  = (const int*)d_in[13];
  float* out = (float*)d_out;

  float* Pq = (float*)d_ws;                       // [8][2048][128] = 8 MB
  float* Pk = Pq + (size_t)BB * NN * DD;          // [8][2048][128] = 8 MB

  proj_kernel<<<(BB * NN) / 256, 256, 0, stream>>>(
      x, emb, gam, bet, w1, w2, gb, wq, bq, wk, bk, inc, Pq, Pk);

  attn_topk_kernel<<<BB * (NN / 16), 256, 0, stream>>>(Pq, Pk, mlpw, mlpb, out);
}